// acLSTM_63230508531714
// MI455X (gfx1250) — compile-verified
//
#include <hip/hip_runtime.h>
#include <hip/hip_bf16.h>

// ---------------------------------------------------------------------------
// 3-layer LSTM decoder (acLSTM), fp32, CDNA5 / gfx1250.
// Matrix math on V_WMMA_F32_16X16X4_F32 (true fp32 tensor op), with
// register-double-buffered global->LDS staging so L2 latency hides under WMMA.
// ---------------------------------------------------------------------------

#define IN_F   171
#define HID    1024
#define OUT_F  171
#define BATCH  128
#define T_COND 50
#define T_GEN  50
#define T_TOT  (T_COND + T_GEN)

typedef float v2f __attribute__((ext_vector_type(2)));
typedef float v8f __attribute__((ext_vector_type(8)));

constexpr int KC  = 32;   // K chunk width
constexpr int LDA = 36;   // padded LDS row stride (floats): 16B-aligned stores,
                          // conflict-free reads (36k mod 64 distinct, k=0..15)

__device__ __forceinline__ float sigmoid_f(float x) {
    return 1.0f / (1.0f + __expf(-x));
}
__device__ __forceinline__ float tanh_f(float x) {
    return 1.0f - 2.0f / (__expf(2.0f * x) + 1.0f);
}

// ---------------------------------------------------------------------------
// One K-segment of the fused gates GEMM, software-pipelined:
//   prefetch chunk k+1 into VGPRs while WMMAs consume chunk k from LDS.
// VEC=true  : K % 32 == 0, rows 16B-aligned -> global_load_b128 staging
// VEC=false : arbitrary K / alignment        -> guarded scalar staging
// ---------------------------------------------------------------------------
template <bool VEC>
__device__ __forceinline__ void gemm_segment(
    v8f acc[4],
    const float* __restrict__ A, long sa,        // activations: rows = batch
    const float* __restrict__ B, long sb,        // weights: rows = gate*H + col
    int K,
    float* ldsA, float* ldsB,
    int tid, int mBase, int n0, int lo, int hi, int wv)
{
    const int r0  = tid >> 3;          // 0..15
    const int kk0 = (tid & 7) << 2;    // 0,4,...,28

    float4 ra[4], rb[4];

    auto prefetch = [&](int kb) {
        #pragma unroll
        for (int i = 0; i < 4; ++i) {
            const int r    = r0 + 16 * i;
            const long arow = (long)(mBase + r) * sa;
            const long brow = (long)((r >> 4) * HID + n0 + (r & 15)) * sb;
            if (VEC) {
                ra[i] = *(const float4*)(A + arow + kb + kk0);
                rb[i] = *(const float4*)(B + brow + kb + kk0);
            } else {
                float av[4], bv[4];
                #pragma unroll
                for (int j = 0; j < 4; ++j) {
                    const int kg = kb + kk0 + j;
                    const bool ok = (kg < K);
                    av[j] = ok ? A[arow + kg] : 0.f;
                    bv[j] = ok ? B[brow + kg] : 0.f;
                }
                ra[i] = float4{av[0], av[1], av[2], av[3]};
                rb[i] = float4{bv[0], bv[1], bv[2], bv[3]};
            }
        }
    };

    prefetch(0);

    for (int kb = 0; kb < K; kb += KC) {
        __syncthreads();   // previous chunk's consumers done -> LDS reusable
        #pragma unroll
        for (int i = 0; i < 4; ++i) {
            const int r = r0 + 16 * i;
            *(float4*)&ldsA[r * LDA + kk0] = ra[i];
            *(float4*)&ldsB[r * LDA + kk0] = rb[i];
        }
        __syncthreads();

        if (kb + KC < K) prefetch(kb + KC);   // loads in flight during WMMAs

        #pragma unroll
        for (int ks = 0; ks < 8; ++ks) {
            const int kc = (ks << 2) + (hi << 1);
            const v2f a = *(const v2f*)&ldsA[(wv * 16 + lo) * LDA + kc];
            #pragma unroll
            for (int g = 0; g < 4; ++g) {
                const v2f b = *(const v2f*)&ldsB[((g << 4) + lo) * LDA + kc];
                acc[g] = __builtin_amdgcn_wmma_f32_16x16x4_f32(
                    false, a, false, b, (short)0, acc[g], false, false);
            }
        }
    }
}

// ---------------------------------------------------------------------------
// Fused LSTM layer step:
//   gates = X @ Wi^T + bi + Hprev @ Wh^T + bh    (M=128, N=4096, K=Kx+1024)
//   c_new = sig(f)*c + sig(i)*tanh(g); h_new = sig(o)*tanh(c_new)
// Grid (HID/16, BATCH/64), block 128 (4 waves). Each wave: 16 rows x 16 cols
// across all 4 gates -> LSTM epilogue entirely in registers.
// ---------------------------------------------------------------------------
__global__ __launch_bounds__(128) void lstm_layer_kernel(
    const float* __restrict__ x, int xStride, int Kx,
    const float* __restrict__ hprev,
    const float* __restrict__ Wi,
    const float* __restrict__ Wh,
    const float* __restrict__ bi,
    const float* __restrict__ bh,
    float* __restrict__ cio,
    float* __restrict__ hout)
{
    __shared__ float ldsA[64 * LDA];
    __shared__ float ldsB[64 * LDA];

    const int tid   = threadIdx.x;
    const int lane  = tid & 31;
    const int lo    = lane & 15;
    const int hi    = lane >> 4;
    const int wv    = tid >> 5;
    const int n0    = blockIdx.x * 16;
    const int mBase = blockIdx.y * 64;

    v8f acc[4];
    #pragma unroll
    for (int g = 0; g < 4; ++g) acc[g] = (v8f){0.f,0.f,0.f,0.f,0.f,0.f,0.f,0.f};

    // Segment 1: X @ Wi^T  (vector path for layers 2/3 where Kx==1024)
    if ((Kx & 31) == 0 && (xStride & 3) == 0) {
        gemm_segment<true >(acc, x, xStride, Wi, Kx, Kx,
                            ldsA, ldsB, tid, mBase, n0, lo, hi, wv);
    } else {
        gemm_segment<false>(acc, x, xStride, Wi, Kx, Kx,
                            ldsA, ldsB, tid, mBase, n0, lo, hi, wv);
    }
    // Segment 2: Hprev @ Wh^T (always aligned, K=1024)
    gemm_segment<true>(acc, hprev, HID, Wh, HID, HID,
                       ldsA, ldsB, tid, mBase, n0, lo, hi, wv);

    // ---- fused bias + gate nonlinearity + state update ----
    const int n = n0 + lo;
    const float bI = bi[0 * HID + n] + bh[0 * HID + n];
    const float bF = bi[1 * HID + n] + bh[1 * HID + n];
    const float bG = bi[2 * HID + n] + bh[2 * HID + n];
    const float bO = bi[3 * HID + n] + bh[3 * HID + n];

    const int mrow = mBase + wv * 16 + (hi << 3);
    #pragma unroll
    for (int r = 0; r < 8; ++r) {
        const int m  = mrow + r;
        const float gi = sigmoid_f(acc[0][r] + bI);
        const float gf = sigmoid_f(acc[1][r] + bF);
        const float gg = tanh_f   (acc[2][r] + bG);
        const float go = sigmoid_f(acc[3][r] + bO);
        const float cp = cio[m * HID + n];
        const float cn = gf * cp + gi * gg;
        cio [m * HID + n] = cn;
        hout[m * HID + n] = go * tanh_f(cn);
    }
}

// ---------------------------------------------------------------------------
// Decoder: out[b, t, :] = h2[b, :] @ W_dec^T + b_dec   (45 MFLOP/step, VALU)
// ---------------------------------------------------------------------------
__global__ __launch_bounds__(256) void decoder_kernel(
    const float* __restrict__ h2,
    const float* __restrict__ Wdec,
    const float* __restrict__ bdec,
    float* __restrict__ out_t)
{
    const int idx = blockIdx.x * blockDim.x + threadIdx.x;
    if (idx >= BATCH * OUT_F) return;
    const int b = idx / OUT_F;
    const int n = idx % OUT_F;

    const float4* hv = (const float4*)(h2 + b * HID);
    const float4* wv = (const float4*)(Wdec + (long)n * HID);
    float s = 0.f;
    #pragma unroll 8
    for (int k = 0; k < HID / 4; ++k) {
        const float4 h = hv[k];
        const float4 w = wv[k];
        s += h.x * w.x + h.y * w.y + h.z * w.z + h.w * w.w;
    }
    out_t[(long)b * (T_TOT * OUT_F) + n] = s + bdec[n];
}

__global__ void zero_kernel(float* __restrict__ p, int n) {
    const int i = blockIdx.x * blockDim.x + threadIdx.x;
    if (i < n) p[i] = 0.f;
}

// ---------------------------------------------------------------------------
extern "C" void kernel_launch(void* const* d_in, const int* in_sizes, int n_in,
                              void* d_out, int out_size, void* d_ws, size_t ws_size,
                              hipStream_t stream) {
    const float* seq  = (const float*)d_in[0];
    const float* Wih1 = (const float*)d_in[1];
    const float* Whh1 = (const float*)d_in[2];
    const float* bih1 = (const float*)d_in[3];
    const float* bhh1 = (const float*)d_in[4];
    const float* Wih2 = (const float*)d_in[5];
    const float* Whh2 = (const float*)d_in[6];
    const float* bih2 = (const float*)d_in[7];
    const float* bhh2 = (const float*)d_in[8];
    const float* Wih3 = (const float*)d_in[9];
    const float* Whh3 = (const float*)d_in[10];
    const float* bih3 = (const float*)d_in[11];
    const float* bhh3 = (const float*)d_in[12];
    const float* Wdec = (const float*)d_in[13];
    const float* bdec = (const float*)d_in[14];
    (void)in_sizes; (void)n_in; (void)out_size; (void)ws_size;

    float* out = (float*)d_out;
    float* ws  = (float*)d_ws;

    const int S = BATCH * HID;
    float* h0[2] = { ws + 0 * S, ws + 1 * S };
    float* h1[2] = { ws + 2 * S, ws + 3 * S };
    float* h2[2] = { ws + 4 * S, ws + 5 * S };
    float* c0 = ws + 6 * S;
    float* c1 = ws + 7 * S;
    float* c2 = ws + 8 * S;

    {
        const int n = 9 * S;
        zero_kernel<<<(n + 255) / 256, 256, 0, stream>>>(ws, n);
    }

    const dim3 gGates(HID / 16, BATCH / 64);
    const dim3 bGates(128);
    const int decBlocks = (BATCH * OUT_F + 255) / 256;

    for (int t = 0; t < T_TOT; ++t) {
        const int rd = t & 1;
        const int wr = rd ^ 1;

        const float* x;
        int sx;
        if (t < T_COND) { x = seq + (long)t * IN_F;        sx = T_COND * IN_F; }
        else            { x = out + (long)(t - 1) * OUT_F; sx = T_TOT * OUT_F; }

        lstm_layer_kernel<<<gGates, bGates, 0, stream>>>(
            x, sx, IN_F, h0[rd], Wih1, Whh1, bih1, bhh1, c0, h0[wr]);
        lstm_layer_kernel<<<gGates, bGates, 0, stream>>>(
            h0[wr], HID, HID, h1[rd], Wih2, Whh2, bih2, bhh2, c1, h1[wr]);
        lstm_layer_kernel<<<gGates, bGates, 0, stream>>>(
            h1[wr], HID, HID, h2[rd], Wih3, Whh3, bih3, bhh3, c2, h2[wr]);
        decoder_kernel<<<decBlocks, 256, 0, stream>>>(
            h2[wr], Wdec, bdec, out + (long)t * OUT_F);
    }
}